// ChangePointKernelAttention_82592221102312
// MI455X (gfx1250) — compile-verified
//
#include <hip/hip_runtime.h>
#include <hip/hip_bf16.h>
#include <math.h>

typedef __attribute__((ext_vector_type(2))) float v2f;
typedef __attribute__((ext_vector_type(8))) float v8f;

#define NSEQ 1024
#define DIM  128

__device__ __forceinline__ v8f wmma_f32(v2f a, v2f b, v8f c) {
  // D = A(16x4 f32) * B(4x16 f32) + C(16x16 f32)
  return __builtin_amdgcn_wmma_f32_16x16x4_f32(false, a, false, b, (short)0, c, false, false);
}

// Raw hardware transcendentals: single v_rcp/v_sqrt/v_rsq instructions instead
// of the IEEE-correct div_scale/fma Newton chains.
__device__ __forceinline__ float frcp_(float x) { return __builtin_amdgcn_rcpf(x); }
__device__ __forceinline__ float fsqrt_(float x) { return __builtin_amdgcn_sqrtf(x); }
__device__ __forceinline__ float frsq_(float x) { return __builtin_amdgcn_rsqf(x); }
__device__ __forceinline__ float sigmoidf_(float x) { return frcp_(1.0f + __expf(-x)); }

// ---------------------------------------------------------------------------
// Pass 1: per-row squared L2 norm and sum of sigmoid(unit vector) over D=128.
// ---------------------------------------------------------------------------
__global__ void __launch_bounds__(256) stats_kernel(const float* __restrict__ X,
                                                    float* __restrict__ n2o,
                                                    float* __restrict__ so,
                                                    int rows) {
  int wave = threadIdx.x >> 5;
  int lane = threadIdx.x & 31;
  int row = blockIdx.x * 8 + wave;
  if (row >= rows) return;
  const float4 xv = *(const float4*)(X + (size_t)row * DIM + lane * 4);
  float s2 = xv.x * xv.x + xv.y * xv.y + xv.z * xv.z + xv.w * xv.w;
#pragma unroll
  for (int off = 16; off >= 1; off >>= 1) s2 += __shfl_xor(s2, off, 32);
  float rs = frsq_(s2);
  float ss = sigmoidf_(xv.x * rs) + sigmoidf_(xv.y * rs) +
             sigmoidf_(xv.z * rs) + sigmoidf_(xv.w * rs);
#pragma unroll
  for (int off = 16; off >= 1; off >>= 1) ss += __shfl_xor(ss, off, 32);
  if (lane == 0) { n2o[row] = s2; so[row] = ss; }
}

// ---------------------------------------------------------------------------
// Pass 2: un-normalized preattn[n,m] for a 16-row stripe of one head, plus
// per-row sum of squares. Three f32 WMMA gram accumulations per 16x16 tile.
// Staging is fully b128 (float4); sigmoids happen in the staging phase so the
// WMMA inner loop is pure ds_load -> v_wmma.
// ---------------------------------------------------------------------------
__global__ void __launch_bounds__(256, 1) preattn_kernel(
    const float* __restrict__ Q, const float* __restrict__ K,
    const float* __restrict__ qn2, const float* __restrict__ kn2,
    const float* __restrict__ sqv, const float* __restrict__ skv,
    float* __restrict__ attn, float* __restrict__ rowss) {
  __shared__ float sQ[16][132];         // query stripe (rows n)
  __shared__ float sK[16][132];         // key stripe (same row indices n)
  __shared__ float sSg[16][132];        // sigmoid(q_unit) stripe
  __shared__ float slabK[8][16][20];    // per-wave B-side key slab (cols m)
  __shared__ float slabQ[8][16][20];    // per-wave B-side query slab (cols m)
  __shared__ float slabS[8][16][20];    // per-wave B-side sigmoid(k_unit) slab
  __shared__ float sRskW[8][8][16];     // per-wave rsqrt(kn2) for its 128 cols
  __shared__ float stQn2[16], stKn2[16], stRsq[16], stSq[16], sRSS[16];

  const int bh = blockIdx.y;
  const int n0 = blockIdx.x * 16;
  const int tid = threadIdx.x;
  const int wave = tid >> 5, lane = tid & 31;
  const int lref = lane & 15, hs = lane >> 4;
  const size_t rowbase = (size_t)bh * NSEQ;

  if (tid < 16) {
    float q2 = qn2[rowbase + n0 + tid];
    stQn2[tid] = q2;
    stRsq[tid] = frsq_(q2);
    stKn2[tid] = kn2[rowbase + n0 + tid];
    stSq[tid] = sqv[rowbase + n0 + tid];
    sRSS[tid] = 0.f;
  }
  // per-wave rsqrt(kn2) table for this wave's 128 columns
#pragma unroll
  for (int jj = 0; jj < 4; ++jj) {
    int idx = lane + 32 * jj;
    sRskW[wave][idx >> 4][idx & 15] = frsq_(kn2[rowbase + wave * 128 + idx]);
  }
  // stripe staging: 16x128, two float4 per thread
#pragma unroll
  for (int j = 0; j < 2; ++j) {
    int g = tid + 256 * j;                  // 0..511
    int r = g >> 5, c4 = (g & 31) * 4;
    *(float4*)&sQ[r][c4] = *(const float4*)&Q[(rowbase + n0 + r) * DIM + c4];
    *(float4*)&sK[r][c4] = *(const float4*)&K[(rowbase + n0 + r) * DIM + c4];
  }
  __syncthreads();
#pragma unroll
  for (int j = 0; j < 2; ++j) {
    int g = tid + 256 * j;
    int r = g >> 5, c4 = (g & 31) * 4;
    float rs = stRsq[r];
    float4 qv = *(const float4*)&sQ[r][c4];
    float4 sv;
    sv.x = sigmoidf_(qv.x * rs);
    sv.y = sigmoidf_(qv.y * rs);
    sv.z = sigmoidf_(qv.z * rs);
    sv.w = sigmoidf_(qv.w * rs);
    *(float4*)&sSg[r][c4] = sv;
  }
  __syncthreads();

  const float inv2sd = 0.044194173824159216f;  // 1/(2*sqrt(128))
  const float invsd = 0.08838834764831843f;    // 1/sqrt(128)
  const float PI_F = 3.14159265358979323846f;
  const float TWOA = 2.0f + 1e-5f;

  float rowacc[8];
#pragma unroll
  for (int r = 0; r < 8; ++r) rowacc[r] = 0.f;

  for (int t = 0; t < 8; ++t) {
    const int m0 = wave * 128 + t * 16;
    const int mcol = m0 + lref;
    const float qn2c = qn2[rowbase + mcol];
    const float kn2c = kn2[rowbase + mcol];
    const float skc = skv[rowbase + mcol];
    const float rskc = sRskW[wave][t][lref];

    v8f cG = {};  // q_n . k_m   (cos_sim after scaling)
    v8f cT = {};  // k_n . q_m   (SE term, transposed gram per torch quirk)
    v8f cP = {};  // sigm(q_unit_n) . sigm(k_unit_m)

    for (int ks = 0; ks < DIM; ks += 16) {
      // slab staging: 16x16 per matrix, two float4 per lane (b128 path)
#pragma unroll
      for (int j = 0; j < 2; ++j) {
        int g = lane + 32 * j;              // 0..63
        int cc = g >> 2, d4 = (g & 3) * 4;
        float4 kv = *(const float4*)&K[(rowbase + m0 + cc) * DIM + ks + d4];
        float4 qv = *(const float4*)&Q[(rowbase + m0 + cc) * DIM + ks + d4];
        float rk = sRskW[wave][t][cc];
        float4 sv;
        sv.x = sigmoidf_(kv.x * rk);
        sv.y = sigmoidf_(kv.y * rk);
        sv.z = sigmoidf_(kv.z * rk);
        sv.w = sigmoidf_(kv.w * rk);
        *(float4*)&slabK[wave][cc][d4] = kv;
        *(float4*)&slabQ[wave][cc][d4] = qv;
        *(float4*)&slabS[wave][cc][d4] = sv;
      }
      // Warm caches for the next slab while the WMMA loop below runs
      // (global_prefetch_b8 on gfx1250).
      if (ks + 16 < DIM) {
        __builtin_prefetch(&K[(rowbase + m0 + (lane >> 1)) * DIM + ks + 16], 0, 1);
        __builtin_prefetch(&Q[(rowbase + m0 + (lane >> 1)) * DIM + ks + 16], 0, 1);
      }
      __syncthreads();
#pragma unroll
      for (int kk = 0; kk < 16; kk += 4) {
        const int d0 = kk + 2 * hs;  // K index of this lane-half within step
        v2f aG = {sQ[lref][ks + d0], sQ[lref][ks + d0 + 1]};
        v2f aT = {sK[lref][ks + d0], sK[lref][ks + d0 + 1]};
        v2f aP = {sSg[lref][ks + d0], sSg[lref][ks + d0 + 1]};
        v2f bG = {slabK[wave][lref][d0], slabK[wave][lref][d0 + 1]};
        v2f bT = {slabQ[wave][lref][d0], slabQ[wave][lref][d0 + 1]};
        v2f bP = {slabS[wave][lref][d0], slabS[wave][lref][d0 + 1]};
        cG = wmma_f32(aG, bG, cG);
        cT = wmma_f32(aT, bT, cT);
        cP = wmma_f32(aP, bP, cP);
      }
      __syncthreads();
    }

#pragma unroll
    for (int r = 0; r < 8; ++r) {
      const int M = r + 8 * hs;
      float tv = cT[r];
      float se = fmaxf(stKn2[M] + qn2c - 2.0f * tv, 0.0f);
      float e1 = __expf(-fsqrt_(se) * inv2sd);
      float g = cG[r] * stRsq[M] * rskc;
      float pres = PI_F * fsqrt_(fmaxf(TWOA - 2.0f * g, 0.0f));
      float sn = __sinf(pres);
      float e2 = __expf(-2.0f * sn * sn * invsd);
      float p = cP[r];
      float qt = (float)DIM - stSq[M] - skc + p;  // sum (1-sq)(1-sk)
      float pre = p * e1 + qt * e2 + (stQn2[M] + kn2c) * inv2sd;
      attn[(rowbase + n0 + M) * (size_t)NSEQ + mcol] = pre;
      rowacc[r] += pre * pre;
    }
  }

#pragma unroll
  for (int r = 0; r < 8; ++r) {
    float v = rowacc[r];
    v += __shfl_xor(v, 1, 32);
    v += __shfl_xor(v, 2, 32);
    v += __shfl_xor(v, 4, 32);
    v += __shfl_xor(v, 8, 32);
    if (lref == 0) atomicAdd(&sRSS[r + 8 * hs], v);
  }
  __syncthreads();
  if (tid < 16) rowss[rowbase + n0 + tid] = sRSS[tid];
}

// ---------------------------------------------------------------------------
// Pass 3: normalize attn rows (L2), rewrite attn, and out = attn @ V via WMMA.
// Stripe processed in two 512-column phases to keep static LDS < 64 KB.
// ---------------------------------------------------------------------------
__global__ void __launch_bounds__(256, 1) outmul_kernel(
    float* __restrict__ attn, const float* __restrict__ V,
    const float* __restrict__ rowss, float* __restrict__ out) {
  __shared__ float sA[16][516];        // half stripe of normalized attn
  __shared__ float slabV[8][16][20];   // per-wave V slab
  __shared__ float stInv[16];

  const int bh = blockIdx.y;
  const int n0 = blockIdx.x * 16;
  const int tid = threadIdx.x;
  const int wave = tid >> 5, lane = tid & 31;
  const int lref = lane & 15, hs = lane >> 4;
  const size_t rowbase = (size_t)bh * NSEQ;

  if (tid < 16) {
    float d = fsqrt_(rowss[rowbase + n0 + tid]);
    stInv[tid] = frcp_(fmaxf(d, 1e-12f));
  }
  __syncthreads();

  const int dcol0 = wave * 16;
  v8f c = {};

  for (int p = 0; p < 2; ++p) {
    const int c0 = p * 512;
    // load + normalize + write back this half of the stripe (float4 path)
#pragma unroll
    for (int j = 0; j < 8; ++j) {
      int g = tid + 256 * j;                 // 0..2047
      int r = g >> 7, c4 = (g & 127) * 4;
      size_t gi = (rowbase + n0 + r) * (size_t)NSEQ + c0 + c4;
      float s = stInv[r];
      float4 av = *(float4*)&attn[gi];
      av.x *= s; av.y *= s; av.z *= s; av.w *= s;
      *(float4*)&sA[r][c4] = av;
      *(float4*)&attn[gi] = av;
    }
    __syncthreads();

    for (int ks = 0; ks < 512; ks += 16) {
#pragma unroll
      for (int j = 0; j < 2; ++j) {
        int g = lane + 32 * j;               // 0..63
        int kr = g >> 2, d4 = (g & 3) * 4;
        *(float4*)&slabV[wave][kr][d4] =
            *(const float4*)&V[(rowbase + c0 + ks + kr) * DIM + dcol0 + d4];
      }
      if (ks + 16 < 512)
        __builtin_prefetch(&V[(rowbase + c0 + ks + 16 + (lane >> 1)) * DIM + dcol0], 0, 1);
      __syncthreads();
#pragma unroll
      for (int kk = 0; kk < 16; kk += 4) {
        const int d0 = kk + 2 * hs;
        v2f a = {sA[lref][ks + d0], sA[lref][ks + d0 + 1]};
        v2f b = {slabV[wave][d0][lref], slabV[wave][d0 + 1][lref]};
        c = wmma_f32(a, b, c);
      }
      __syncthreads();
    }
    __syncthreads();
  }

#pragma unroll
  for (int r = 0; r < 8; ++r)
    out[(rowbase + n0 + r + 8 * hs) * DIM + dcol0 + lref] = c[r];
}

// ---------------------------------------------------------------------------
extern "C" void kernel_launch(void* const* d_in, const int* in_sizes, int n_in,
                              void* d_out, int out_size, void* d_ws, size_t ws_size,
                              hipStream_t stream) {
  (void)n_in; (void)out_size; (void)ws_size;
  const int N = NSEQ, D = DIM;
  const float* q = (const float*)d_in[0];
  const float* k = (const float*)d_in[1];
  const float* v = (const float*)d_in[2];
  const int BH = in_sizes[0] / (N * D);   // 64
  const int BHN = BH * N;                 // 65536

  float* outp = (float*)d_out;                       // [BH, N, D]
  float* attnp = outp + (size_t)BH * N * D;          // [BH, N, N]

  float* qn2 = (float*)d_ws;       // workspace: 5 * BHN floats = 1.25 MB
  float* kn2 = qn2 + BHN;
  float* sqs = kn2 + BHN;
  float* sks = sqs + BHN;
  float* rss = sks + BHN;

  dim3 blk(256);
  stats_kernel<<<dim3((BHN + 7) / 8), blk, 0, stream>>>(q, qn2, sqs, BHN);
  stats_kernel<<<dim3((BHN + 7) / 8), blk, 0, stream>>>(k, kn2, sks, BHN);
  preattn_kernel<<<dim3(N / 16, BH), blk, 0, stream>>>(q, k, qn2, kn2, sqs, sks,
                                                       attnp, rss);
  outmul_kernel<<<dim3(N / 16, BH), blk, 0, stream>>>(attnp, v, rss, outp);
}